// DensityGrid_16647293239639
// MI455X (gfx1250) — compile-verified
//
#include <hip/hip_runtime.h>
#include <hip/hip_bf16.h>

// ---------------------------------------------------------------------------
// DensityGrid on MI455X (gfx1250, wave32).
// Hot loop: 288x fused 3x3x3 maxpool-and-mask over a 96^3 grid (CCL by
// dilation). 3.5MB dataset -> L2 resident; fused LDS-tiled stencil, tile
// fetched via gfx1250 async LDS-DMA (GLOBAL_LOAD_ASYNC_TO_LDS_B128 +
// s_wait_asynccnt). k_ccl_pool is defined first so the disasm snippet shows
// the async tile-load path.
// ---------------------------------------------------------------------------

#define G 96
#define N3 (G * G * G)          // 884736
#define CCL_ITERS (3 * G)       // 288
#define TPB 256

// ---- gfx1250 async global->LDS copy helpers -------------------------------
#if defined(__HIP_DEVICE_COMPILE__) && __has_builtin(__builtin_amdgcn_global_load_async_to_lds_b128)
#define USE_ASYNC_LDS 1
#else
#define USE_ASYNC_LDS 0
#endif

#if USE_ASYNC_LDS
typedef int v4i_t __attribute__((vector_size(16)));
typedef __attribute__((address_space(1))) v4i_t gv4i_t;   // global
typedef __attribute__((address_space(3))) v4i_t lv4i_t;   // LDS
#endif

__device__ __forceinline__ void tile_copy16(const float* g, void* ldsDst) {
#if USE_ASYNC_LDS
    __builtin_amdgcn_global_load_async_to_lds_b128(
        (gv4i_t*)g, (lv4i_t*)ldsDst, /*offset=*/0, /*cpol=*/0);
#else
    *(float4*)ldsDst = *(const float4*)g;
#endif
}

__device__ __forceinline__ void tile_copy_wait() {
#if USE_ASYNC_LDS
#if __has_builtin(__builtin_amdgcn_s_wait_asynccnt)
    __builtin_amdgcn_s_wait_asynccnt(0);
#else
    asm volatile("s_wait_asynccnt 0" ::: "memory");
#endif
#endif
}

// ---- fused separable 3x3x3 maxpool core (stride1 pad1, values >= 0) --------
// Block tile: full-x rows, 4x4 (y,z) interior, +1 halo -> 6x6 rows of 96.
// Loads tile into s_in (async DMA), x-pools into s_xp. After return the
// caller reads the 9-point y/z max from s_xp.
__device__ __forceinline__ void pool_tile_load_and_xpool(
    const float* __restrict__ in, float* s_in, float* s_xp,
    int y0, int z0, int tid) {
    // Load 36 rows x 96 floats = 864 x 16B segments (async DMA to LDS).
    for (int t = tid; t < 36 * 24; t += TPB) {
        int row = t / 24, seg = t % 24;
        int dz = row / 6, dy = row % 6;
        int z = z0 - 1 + dz, y = y0 - 1 + dy;
        char* dst = (char*)s_in + (size_t)(row * G + seg * 4) * 4;
        if ((unsigned)z < (unsigned)G && (unsigned)y < (unsigned)G) {
            tile_copy16(in + ((z * G + y) * G + seg * 4), dst);
        } else {
            float4 zero = {0.f, 0.f, 0.f, 0.f};   // zero pad == -inf pad (vals >= 0)
            *(float4*)dst = zero;
        }
    }
    tile_copy_wait();
    __syncthreads();

    // x-direction pool in LDS
    for (int t = tid; t < 36 * G; t += TPB) {
        int x = t % G, r = t / G;
        float c = s_in[r * G + x];
        float l = (x > 0)     ? s_in[r * G + x - 1] : 0.0f;
        float h = (x < G - 1) ? s_in[r * G + x + 1] : 0.0f;
        s_xp[r * G + x] = fmaxf(c, fmaxf(l, h));
    }
    __syncthreads();
}

__device__ __forceinline__ float pool_yz_max(const float* s_xp, int zz, int yy, int x) {
    int base = (zz * 6 + yy) * G + x;   // dz-1=zz, dy-1=yy
    float v = 0.0f;
#pragma unroll
    for (int a = 0; a < 3; ++a)
#pragma unroll
        for (int b = 0; b < 3; ++b)
            v = fmaxf(v, s_xp[base + a * 6 * G + b * G]);
    return v;
}

// ---- CCL step: out = maxpool3d(in) * mask (executed 288x) ------------------
__global__ void k_ccl_pool(const float* __restrict__ in,
                           float* __restrict__ out,
                           const unsigned char* __restrict__ mask) {
    __shared__ float s_in[36 * G];
    __shared__ float s_xp[36 * G];
    const int tid = threadIdx.x;
    const int y0  = blockIdx.x * 4;
    const int z0  = blockIdx.y * 4;

    if (tid < 16) {     // prefetch interior mask rows (gfx1250 global_prefetch_b8)
        int yy = tid & 3, zz = tid >> 2;
        __builtin_prefetch(&mask[((z0 + zz) * G + (y0 + yy)) * G], 0, 3);
    }

    pool_tile_load_and_xpool(in, s_in, s_xp, y0, z0, tid);

    for (int t = tid; t < 16 * G; t += TPB) {
        int x  = t % G;
        int yy = (t / G) & 3;
        int zz = t / (4 * G);
        float v = pool_yz_max(s_xp, zz, yy, x);
        int gi = ((z0 + zz) * G + (y0 + yy)) * G + x;
        out[gi] = mask[gi] ? v : 0.0f;
    }
}

// ---- field pool: out = maxpool3d(in), plus global sum for the mean ---------
__global__ void k_field_pool(const float* __restrict__ in,
                             float* __restrict__ out,
                             float* __restrict__ sumbuf) {
    __shared__ float s_in[36 * G];
    __shared__ float s_xp[36 * G];
    __shared__ float red[TPB];
    const int tid = threadIdx.x;
    const int y0  = blockIdx.x * 4;
    const int z0  = blockIdx.y * 4;

    pool_tile_load_and_xpool(in, s_in, s_xp, y0, z0, tid);

    float lsum = 0.0f;
    for (int t = tid; t < 16 * G; t += TPB) {
        int x  = t % G;
        int yy = (t / G) & 3;
        int zz = t / (4 * G);
        float v = pool_yz_max(s_xp, zz, yy, x);
        int gi = ((z0 + zz) * G + (y0 + yy)) * G + x;
        out[gi] = v;
        lsum += v;
    }

    red[tid] = lsum;
    __syncthreads();
    for (int s = TPB / 2; s > 0; s >>= 1) {
        if (tid < s) red[tid] += red[tid + s];
        __syncthreads();
    }
    if (tid == 0) atomicAdd(sumbuf, red[0]);
}

// ---- elementwise prologue --------------------------------------------------
// out_density = 1-exp(-0.01*relu(d)); new_cached = max(0.8*dc, relu(d));
// field0 = 1-exp(-0.01*new_cached)
__global__ void k_prep(const float* __restrict__ den,
                       const float* __restrict__ dc,
                       float* __restrict__ out_density,
                       float* __restrict__ new_cached,
                       float* __restrict__ field0) {
    for (int i = blockIdx.x * blockDim.x + threadIdx.x; i < N3;
         i += gridDim.x * blockDim.x) {
        float d    = den[i];
        float dpos = fmaxf(d, 0.0f);
        float nc   = fmaxf(dc[i] * 0.8f, dpos);
        new_cached[i]  = nc;
        out_density[i] = 1.0f - __expf(-0.01f * dpos);
        field0[i]      = 1.0f - __expf(-0.01f * nc);
    }
}

// ---- threshold + label init ------------------------------------------------
__global__ void k_mask_comp(const float* __restrict__ pooled,
                            const float* __restrict__ sumbuf,
                            unsigned char* __restrict__ mask,
                            float* __restrict__ comp) {
    const float thr = fminf(sumbuf[0] * (1.0f / (float)N3), 0.01f);
    for (int i = blockIdx.x * blockDim.x + threadIdx.x; i < N3;
         i += gridDim.x * blockDim.x) {
        bool m = pooled[i] > thr;
        mask[i] = m ? 1u : 0u;
        comp[i] = m ? (float)(i + 1) : 0.0f;   // labels < 2^24, exact in f32
    }
}

// ---- histogram of labels over mask ----------------------------------------
__global__ void k_hist(const float* __restrict__ comp,
                       const unsigned char* __restrict__ mask,
                       unsigned* __restrict__ counts) {
    for (int i = blockIdx.x * blockDim.x + threadIdx.x; i < N3;
         i += gridDim.x * blockDim.x) {
        if (mask[i]) atomicAdd(&counts[(unsigned)comp[i]], 1u);
    }
}

// ---- argmax with smallest-label tie-break (labels 1..N3) -------------------
// key = (count << 32) | (0xFFFFFFFF - label): max key == max count, then min
// label. Matches jnp.argmax over counts with counts[0] = -1.
__global__ void k_argmax(const unsigned* __restrict__ counts,
                         unsigned long long* __restrict__ keyp) {
    __shared__ unsigned long long sred[TPB];
    unsigned long long best = 0ull;
    for (int i = 1 + blockIdx.x * blockDim.x + threadIdx.x; i <= N3;
         i += gridDim.x * blockDim.x) {
        unsigned long long k =
            ((unsigned long long)counts[i] << 32) | (0xFFFFFFFFu - (unsigned)i);
        best = (k > best) ? k : best;
    }
    sred[threadIdx.x] = best;
    __syncthreads();
    for (int s = TPB / 2; s > 0; s >>= 1) {
        if (threadIdx.x < s)
            sred[threadIdx.x] = max(sred[threadIdx.x], sred[threadIdx.x + s]);
        __syncthreads();
    }
    if (threadIdx.x == 0) atomicMax(keyp, sred[0]);
}

// ---- epilogue: new_field + valid ------------------------------------------
__global__ void k_final(const float* __restrict__ comp,
                        const unsigned long long* __restrict__ keyp,
                        const unsigned char* __restrict__ old_field,
                        const int* __restrict__ step,
                        float* __restrict__ valid_out,
                        float* __restrict__ nf_out) {
    const unsigned label = 0xFFFFFFFFu - (unsigned)(*keyp & 0xFFFFFFFFull);
    const int s = *step;
    for (int i = blockIdx.x * blockDim.x + threadIdx.x; i < N3;
         i += gridDim.x * blockDim.x) {
        float nf = ((unsigned)comp[i] == label) ? 1.0f : 0.0f;
        nf_out[i]    = nf;
        valid_out[i] = (s < 500) ? nf : (old_field[i] ? 1.0f : 0.0f);
    }
}

// ---------------------------------------------------------------------------
extern "C" void kernel_launch(void* const* d_in, const int* in_sizes, int n_in,
                              void* d_out, int out_size, void* d_ws, size_t ws_size,
                              hipStream_t stream) {
    (void)in_sizes; (void)n_in; (void)out_size; (void)ws_size;

    const float*         density = (const float*)d_in[0];
    const float*         cached  = (const float*)d_in[1];
    const unsigned char* oldf    = (const unsigned char*)d_in[2]; // jax bool = 1B
    const int*           step    = (const int*)d_in[3];

    float* out_density = (float*)d_out;               // [0, N3)
    float* out_valid   = (float*)d_out + N3;          // [N3, 2N3)
    float* out_nfield  = (float*)d_out + 2 * N3;      // [2N3, 3N3)
    float* out_cached  = (float*)d_out + 3 * N3;      // [3N3, 4N3)

    // workspace layout
    char* ws = (char*)d_ws;
    float*              bufA   = (float*)(ws);                       // 4*N3
    float*              bufB   = (float*)(ws + 4ull * N3);           // 4*N3
    unsigned char*      mask   = (unsigned char*)(ws + 8ull * N3);   // N3
    unsigned*           counts = (unsigned*)(ws + 9ull * N3);        // 4*(N3+1)
    float*              sumb   = (float*)(ws + 13ull * N3 + 4);      // 4
    unsigned long long* keyp   = (unsigned long long*)(ws + 13ull * N3 + 8); // 8

    // zero counts + sum + key (contiguous region), capturable stream op
    (void)hipMemsetAsync(counts, 0, 4ull * N3 + 16, stream);

    const int blocks = (N3 + TPB - 1) / TPB;
    const dim3 pgrid(G / 4, G / 4);   // 24 x 24 tiles

    // 1) elementwise prologue; field0 -> bufA
    k_prep<<<blocks, TPB, 0, stream>>>(density, cached, out_density, out_cached, bufA);

    // 2) pooled field -> bufB, plus sum for the mean
    k_field_pool<<<pgrid, TPB, 0, stream>>>(bufA, bufB, sumb);

    // 3) mask + initial labels -> bufA
    k_mask_comp<<<blocks, TPB, 0, stream>>>(bufB, sumb, mask, bufA);

    // 4) 288x dilate-and-mask (ping-pong; even count -> result back in bufA)
    float* src = bufA;
    float* dst = bufB;
    for (int it = 0; it < CCL_ITERS; ++it) {
        k_ccl_pool<<<pgrid, TPB, 0, stream>>>(src, dst, mask);
        float* t = src; src = dst; dst = t;
    }

    // 5) histogram + argmax(first-max) + epilogue
    k_hist<<<blocks, TPB, 0, stream>>>(src, mask, counts);
    k_argmax<<<512, TPB, 0, stream>>>(counts, keyp);
    k_final<<<blocks, TPB, 0, stream>>>(src, keyp, oldf, step, out_valid, out_nfield);
}